// LFQQuantizer_21895743275554
// MI455X (gfx1250) — compile-verified
//
#include <hip/hip_runtime.h>

typedef __attribute__((ext_vector_type(2))) float v2f;
typedef __attribute__((ext_vector_type(8))) float v8f;

#define NUM_CODES 1024
#define CODE_DIM  64
#define BATCH     4096

// One wave handles 16 rows of z_e. Block = 256 threads = 8 waves = 128 rows.
__global__ __launch_bounds__(256) void lfq_nn_wmma_kernel(
    const float* __restrict__ z_e,
    const float* __restrict__ codebook,
    float* __restrict__ z_q,
    float* __restrict__ out_idx) {
  const int lane = threadIdx.x & 31;
  const int wave = threadIdx.x >> 5;
  const int rowBase = (blockIdx.x * 8 + wave) * 16;
  const int half = lane >> 4;   // 0: C rows 0-7, A/B K-pair {0,1}; 1: rows 8-15, K-pair {2,3}
  const int l16  = lane & 15;   // A row % 16 / B column % 16

  // ---- Load A fragments: 16 K-slabs of width 4 covering dim 64 ----
  // 32-bit A 16x4 layout: lane L holds row M=L%16; VGPR0/1 hold K = (L<16 ? {0,1} : {2,3})
  v2f a[16];
  const float* zrow = z_e + (size_t)(rowBase + l16) * CODE_DIM;
#pragma unroll
  for (int s = 0; s < 16; ++s) {
    const int kb = 4 * s + 2 * half;
    a[s] = *(const v2f*)(zrow + kb);
  }

  float best[8];
  int   bidx[8];
#pragma unroll
  for (int r = 0; r < 8; ++r) { best[r] = 3.4e38f; bidx[r] = 0; }

  // ---- Sweep all 1024 codes in 16-column tiles ----
  for (int n0 = 0; n0 < NUM_CODES; n0 += 16) {
    const int col = n0 + l16;
    const float* crow = codebook + (size_t)col * CODE_DIM;

    // B fragments (4x16 per slab): lane holds column N=lane%16, K striping mirrors A.
    v2f b[16];
    float pn = 0.0f;
#pragma unroll
    for (int s = 0; s < 16; ++s) {
      const int kb = 4 * s + 2 * half;
      b[s] = *(const v2f*)(crow + kb);
      pn += b[s].x * b[s].x + b[s].y * b[s].y;
    }
    // ||c||^2 for this lane's column: this lane saw half the K range, partner lane^16 the rest.
    const float cnorm = pn + __shfl_xor(pn, 16, 32);

    // acc[r] = dot(z_row(8*half + r), code_col(l16)), exact f32 via V_WMMA_F32_16X16X4_F32
    v8f acc = {};
#pragma unroll
    for (int s = 0; s < 16; ++s) {
      acc = __builtin_amdgcn_wmma_f32_16x16x4_f32(
          /*neg_a=*/false, a[s], /*neg_b=*/false, b[s],
          /*c_mod=*/(short)0, acc, /*reuse_a=*/false, /*reuse_b=*/false);
    }

    // score = ||c||^2 - 2 z.c  (||z||^2 dropped: constant per row, argmin-invariant)
#pragma unroll
    for (int r = 0; r < 8; ++r) {
      const float score = __builtin_fmaf(-2.0f, acc[r], cnorm);
      if (score < best[r]) { best[r] = score; bidx[r] = col; }
    }
  }

  // ---- Min+argmin across the 16 lanes of each half (ties -> lower index) ----
#pragma unroll
  for (int off = 1; off < 16; off <<= 1) {
#pragma unroll
    for (int r = 0; r < 8; ++r) {
      const float ob = __shfl_xor(best[r], off, 32);
      const int   oi = __shfl_xor(bidx[r], off, 32);
      if (ob < best[r] || (ob == best[r] && oi < bidx[r])) { best[r] = ob; bidx[r] = oi; }
    }
  }

  // ---- Write indices (one lane per half covers its 8 rows) ----
  if (l16 == 0) {
#pragma unroll
    for (int r = 0; r < 8; ++r) {
      out_idx[rowBase + half * 8 + r] = (float)bidx[r];
    }
  }

  // ---- Gather z_q = codebook[idx]: 16 lanes/half x float4 = 64 floats per row ----
#pragma unroll
  for (int r = 0; r < 8; ++r) {
    const int row = rowBase + half * 8 + r;
    const int idx = bidx[r];  // uniform within the half after reduction
    const float4 v = *(const float4*)(codebook + (size_t)idx * CODE_DIM + l16 * 4);
    *(float4*)(z_q + (size_t)row * CODE_DIM + l16 * 4) = v;
  }
}

extern "C" void kernel_launch(void* const* d_in, const int* in_sizes, int n_in,
                              void* d_out, int out_size, void* d_ws, size_t ws_size,
                              hipStream_t stream) {
  const float* z_e      = (const float*)d_in[0];
  const float* codebook = (const float*)d_in[1];
  float* z_q  = (float*)d_out;                       // (4096, 64) f32
  float* idxs = (float*)d_out + BATCH * CODE_DIM;    // (4096,) indices, stored as float
  lfq_nn_wmma_kernel<<<BATCH / 128, 256, 0, stream>>>(z_e, codebook, z_q, idxs);
}